// Spatial_Conv_463856468395
// MI455X (gfx1250) — compile-verified
//
#include <hip/hip_runtime.h>
#include <math.h>

#define CH   64
#define TILE 32      // nodes per block tile
#define XS   68      // padded LDS row stride (floats) -> conflict-free A-frag reads

typedef __attribute__((ext_vector_type(2))) float v2f;
typedef __attribute__((ext_vector_type(8))) float v8f;
typedef __attribute__((ext_vector_type(4))) int   v4i;

#define AS1 __attribute__((address_space(1)))
#define AS3 __attribute__((address_space(3)))

#if __has_builtin(__builtin_amdgcn_global_load_async_to_lds_b128)
#define HAVE_ASYNC_LDS 1
#else
#define HAVE_ASYNC_LDS 0
#endif

// 16B global -> LDS copy. Async (ASYNCcnt-tracked, no VGPR staging) when the
// gfx1250 builtin is available; plain copy otherwise.
// Signature (from compiler diagnostic): param0 = AS1 v4i* (global src),
// param1 = LDS dst, then imm offset, imm cpol.
__device__ __forceinline__ void copy_b128_g2l(const float* g, float* l) {
#if HAVE_ASYNC_LDS
    __builtin_amdgcn_global_load_async_to_lds_b128(
        (AS1 v4i*)(AS1 void*)(void*)g,
        (AS3 v4i*)(AS3 void*)(void*)l, 0, 0);
#else
    l[0] = g[0]; l[1] = g[1]; l[2] = g[2]; l[3] = g[3];
#endif
}
__device__ __forceinline__ void async_fill_wait() {
#if HAVE_ASYNC_LDS
# if __has_builtin(__builtin_amdgcn_s_wait_asynccnt)
    __builtin_amdgcn_s_wait_asynccnt(0);
# else
    asm volatile("s_wait_asynccnt 0" ::: "memory");
# endif
#endif
}

__device__ __forceinline__ float sigmoid_f(float x) {
    return 1.0f / (1.0f + expf(-x));
}
__device__ __forceinline__ float gelu_f(float x) {
    return 0.5f * x * (1.0f + erff(x * 0.70710678118654752f));
}

// One 16(nodes) x 16(cols) output tile of  X[16x64] @ W[64x64], K-loop via
// 16 chained V_WMMA_F32_16X16X4_F32.
//   A 16x4 f32 layout:  lane<16: M=lane, v0=K0,v1=K1 ; lane>=16: M=lane-16, v0=K2,v1=K3
//   B 4x16  f32 layout:  lane<16: N=lane, v0=K0,v1=K1 ; lane>=16: N=lane-16, v0=K2,v1=K3
__device__ __forceinline__ v8f wmma_tile(const float* __restrict__ sX,
                                         const float* __restrict__ sW,
                                         int g, int lane)
{
    v8f acc = {0.f, 0.f, 0.f, 0.f, 0.f, 0.f, 0.f, 0.f};
    const int m  = lane & 15;
    const int kb = (lane >> 4) << 1;
    const int n  = (g << 4) + (lane & 15);
#pragma unroll
    for (int k0 = 0; k0 < CH; k0 += 4) {
        v2f a, b;
        a.x = sX[m * XS + k0 + kb];
        a.y = sX[m * XS + k0 + kb + 1];
        b.x = sW[(k0 + kb) * CH + n];
        b.y = sW[(k0 + kb + 1) * CH + n];
        acc = __builtin_amdgcn_wmma_f32_16x16x4_f32(false, a, false, b,
                                                    (short)0, acc, false, false);
    }
    return acc;
}

// ---------------- pre: gates, h1 = gelu((ig*x)@W_in+b), m = h1@W_arma --------
__global__ __launch_bounds__(256) void pre_kernel(
    const float* __restrict__ x,
    const float* __restrict__ Wig, const float* __restrict__ big,
    const float* __restrict__ Wog, const float* __restrict__ bog,
    const float* __restrict__ Win, const float* __restrict__ bin,
    const float* __restrict__ War,
    float* __restrict__ og, float* __restrict__ h1, float* __restrict__ mb,
    int nNodes, int nTiles)
{
    __shared__ float sWig[CH * CH], sWog[CH * CH], sWin[CH * CH], sWar[CH * CH];
    __shared__ float sbig[CH], sbog[CH], sbin[CH];
    __shared__ float sX[TILE * XS], sH[TILE * XS];

    // async weight fill (16KB x4), waited at the first in-loop async_fill_wait
    for (int i = threadIdx.x; i < (CH * CH) / 4; i += blockDim.x) {
        copy_b128_g2l(&Wig[i * 4], &sWig[i * 4]);
        copy_b128_g2l(&Wog[i * 4], &sWog[i * 4]);
        copy_b128_g2l(&Win[i * 4], &sWin[i * 4]);
        copy_b128_g2l(&War[i * 4], &sWar[i * 4]);
    }
    if (threadIdx.x < CH) {
        sbig[threadIdx.x] = big[threadIdx.x];
        sbog[threadIdx.x] = bog[threadIdx.x];
        sbin[threadIdx.x] = bin[threadIdx.x];
    }

    const int lane  = threadIdx.x & 31;
    const int w     = threadIdx.x >> 5;
    const int s     = w >> 2;             // node sub-tile (0..1)
    const int g     = w & 3;              // 16-col group (0..3)
    const int col   = (g << 4) + (lane & 15);
    const int rbase = (s << 4) + ((lane >> 4) << 3);

    for (int tile = blockIdx.x; tile < nTiles; tile += gridDim.x) {
        const int base = tile * TILE;
        __syncthreads();                  // prev iteration done reading sX
        for (int i = threadIdx.x; i < TILE * (CH / 4); i += blockDim.x) {
            int r = i >> 4, c = (i & 15) << 2;
            int node = base + r;
            float* dst = &sX[r * XS + c];
            if (node < nNodes) copy_b128_g2l(&x[(size_t)node * CH + c], dst);
            else { dst[0] = 0.f; dst[1] = 0.f; dst[2] = 0.f; dst[3] = 0.f; }
        }
        async_fill_wait();
        __syncthreads();

        const float* sXs = &sX[(s << 4) * XS];
        v8f aig = wmma_tile(sXs, sWig, g, lane);
        v8f aog = wmma_tile(sXs, sWog, g, lane);
#pragma unroll
        for (int r = 0; r < 8; ++r) {
            int row = rbase + r, node = base + row;
            float igv = sigmoid_f(aig[r] + sbig[col]);
            float ogv = sigmoid_f(aog[r] + sbog[col]);
            sH[row * XS + col] = igv * sX[row * XS + col];
            if (node < nNodes) og[(size_t)node * CH + col] = ogv;
        }
        __syncthreads();

        v8f a1 = wmma_tile(&sH[(s << 4) * XS], sWin, g, lane);
#pragma unroll
        for (int r = 0; r < 8; ++r) {
            int row = rbase + r, node = base + row;
            float h = gelu_f(a1[r] + sbin[col]);
            sX[row * XS + col] = h;
            if (node < nNodes) h1[(size_t)node * CH + col] = h;
        }
        __syncthreads();

        v8f am = wmma_tile(&sX[(s << 4) * XS], sWar, g, lane);
#pragma unroll
        for (int r = 0; r < 8; ++r) {
            int row = rbase + r, node = base + row;
            if (node < nNodes) mb[(size_t)node * CH + col] = am[r];
        }
    }
}

// ---------------- edge normalization + scatter -------------------------------
__global__ void zero_kernel(float* __restrict__ p, long long n) {
    long long i = (long long)blockIdx.x * blockDim.x + threadIdx.x;
    if (i < n) p[i] = 0.0f;
}

__global__ void deg_kernel(const int* __restrict__ colv, float* __restrict__ deg, int E) {
    int e = blockIdx.x * blockDim.x + threadIdx.x;
    if (e < E)
        __hip_atomic_fetch_add(&deg[colv[e]], 1.0f, __ATOMIC_RELAXED,
                               __HIP_MEMORY_SCOPE_AGENT);
}

__global__ void dis_kernel(float* __restrict__ deg, int nNodes) {
    int i = blockIdx.x * blockDim.x + threadIdx.x;
    if (i < nNodes) {
        float d = deg[i];
        deg[i] = (d > 0.0f) ? rsqrtf(fmaxf(d, 1.0f)) : 0.0f;
    }
}

__global__ void norm_kernel(const int* __restrict__ rowv, const int* __restrict__ colv,
                            const float* __restrict__ dis, float* __restrict__ nrm, int E) {
    int e = blockIdx.x * blockDim.x + threadIdx.x;
    if (e < E) nrm[e] = dis[rowv[e]] * dis[colv[e]];
}

// thread per (edge, channel): coalesced gather of m[row], coalesced fp32 atomic
// into agg[col] (L2-resident: m + agg = 51MB << 192MB L2).
__global__ void scatter_kernel(const int* __restrict__ rowv, const int* __restrict__ colv,
                               const float* __restrict__ nrm, const float* __restrict__ mb,
                               float* __restrict__ agg, long long total) {
    long long idx = (long long)blockIdx.x * blockDim.x + threadIdx.x;
    if (idx >= total) return;
    int e = (int)(idx >> 6);
    int c = (int)(idx & 63);
    float v = nrm[e] * mb[(size_t)rowv[e] * CH + c];
    __hip_atomic_fetch_add(&agg[(size_t)colv[e] * CH + c], v, __ATOMIC_RELAXED,
                           __HIP_MEMORY_SCOPE_AGENT);
}

// ---------------- post: relu(agg + h1@V+b) -> gelu(@W_gw+b) -> og*(@W_out+b) -
__global__ __launch_bounds__(256) void post_kernel(
    const float* __restrict__ agg, const float* __restrict__ h1,
    const float* __restrict__ og,
    const float* __restrict__ Var, const float* __restrict__ bar,
    const float* __restrict__ Wgw, const float* __restrict__ bgw,
    const float* __restrict__ Wout, const float* __restrict__ bout,
    float* __restrict__ out, int nNodes, int nTiles)
{
    __shared__ float sVar[CH * CH], sWgw[CH * CH], sWout[CH * CH];
    __shared__ float sbar[CH], sbgw[CH], sbout[CH];
    __shared__ float sA[TILE * XS], sB[TILE * XS];

    for (int i = threadIdx.x; i < (CH * CH) / 4; i += blockDim.x) {
        copy_b128_g2l(&Var[i * 4],  &sVar[i * 4]);
        copy_b128_g2l(&Wgw[i * 4],  &sWgw[i * 4]);
        copy_b128_g2l(&Wout[i * 4], &sWout[i * 4]);
    }
    if (threadIdx.x < CH) {
        sbar[threadIdx.x]  = bar[threadIdx.x];
        sbgw[threadIdx.x]  = bgw[threadIdx.x];
        sbout[threadIdx.x] = bout[threadIdx.x];
    }

    const int lane  = threadIdx.x & 31;
    const int w     = threadIdx.x >> 5;
    const int s     = w >> 2;
    const int g     = w & 3;
    const int col   = (g << 4) + (lane & 15);
    const int rbase = (s << 4) + ((lane >> 4) << 3);

    for (int tile = blockIdx.x; tile < nTiles; tile += gridDim.x) {
        const int base = tile * TILE;
        __syncthreads();
        for (int i = threadIdx.x; i < TILE * (CH / 4); i += blockDim.x) {
            int r = i >> 4, c = (i & 15) << 2;
            int node = base + r;
            float* dA = &sA[r * XS + c];
            float* dB = &sB[r * XS + c];
            if (node < nNodes) {
                copy_b128_g2l(&h1[(size_t)node * CH + c],  dA);
                copy_b128_g2l(&agg[(size_t)node * CH + c], dB);
            } else {
                dA[0] = dA[1] = dA[2] = dA[3] = 0.f;
                dB[0] = dB[1] = dB[2] = dB[3] = 0.f;
            }
        }
        async_fill_wait();
        __syncthreads();

        v8f a0 = wmma_tile(&sA[(s << 4) * XS], sVar, g, lane);
#pragma unroll
        for (int r = 0; r < 8; ++r) {
            int row = rbase + r;
            float h2 = a0[r] + sB[row * XS + col] + sbar[col];
            sB[row * XS + col] = fmaxf(h2, 0.0f);          // relu
        }
        __syncthreads();

        v8f a1 = wmma_tile(&sB[(s << 4) * XS], sWgw, g, lane);
#pragma unroll
        for (int r = 0; r < 8; ++r) {
            int row = rbase + r;
            sA[row * XS + col] = gelu_f(a1[r] + sbgw[col]);
        }
        __syncthreads();

        v8f a2 = wmma_tile(&sA[(s << 4) * XS], sWout, g, lane);
#pragma unroll
        for (int r = 0; r < 8; ++r) {
            int row = rbase + r, node = base + row;
            if (node < nNodes) {
                float o = (a2[r] + sbout[col]) * og[(size_t)node * CH + col];
                out[(size_t)node * CH + col] = o;
            }
        }
    }
}

// ---------------- host launch ------------------------------------------------
extern "C" void kernel_launch(void* const* d_in, const int* in_sizes, int n_in,
                              void* d_out, int out_size, void* d_ws, size_t ws_size,
                              hipStream_t stream)
{
    const float* x    = (const float*)d_in[0];
    const int*   ei   = (const int*)d_in[1];
    const float* Wig  = (const float*)d_in[2];
    const float* big  = (const float*)d_in[3];
    const float* Wog  = (const float*)d_in[4];
    const float* bog  = (const float*)d_in[5];
    const float* Win  = (const float*)d_in[6];
    const float* bin  = (const float*)d_in[7];
    const float* War  = (const float*)d_in[8];
    const float* Var  = (const float*)d_in[9];
    const float* bar  = (const float*)d_in[10];
    const float* Wgw  = (const float*)d_in[11];
    const float* bgw  = (const float*)d_in[12];
    const float* Wout = (const float*)d_in[13];
    const float* bout = (const float*)d_in[14];
    float* out = (float*)d_out;

    const int nNodes = in_sizes[0] / CH;
    const int E      = in_sizes[1] / 2;
    const int* rowv  = ei;        // edge_index[0] : source
    const int* colv  = ei + E;    // edge_index[1] : target

    float* og  = (float*)d_ws;
    float* h1  = og  + (size_t)nNodes * CH;
    float* mb  = h1  + (size_t)nNodes * CH;
    float* agg = mb  + (size_t)nNodes * CH;
    float* deg = agg + (size_t)nNodes * CH;   // becomes dis in-place
    float* nrm = deg + (size_t)nNodes;

    const int nTiles = (nNodes + TILE - 1) / TILE;
    const int gemmGrid = nTiles < 720 ? nTiles : 720;

    long long zt = (long long)nNodes * CH + nNodes;       // agg + deg contiguous
    zero_kernel<<<(int)((zt + 255) / 256), 256, 0, stream>>>(agg, zt);

    pre_kernel<<<gemmGrid, 256, 0, stream>>>(x, Wig, big, Wog, bog, Win, bin, War,
                                             og, h1, mb, nNodes, nTiles);

    deg_kernel<<<(E + 255) / 256, 256, 0, stream>>>(colv, deg, E);
    dis_kernel<<<(nNodes + 255) / 256, 256, 0, stream>>>(deg, nNodes);
    norm_kernel<<<(E + 255) / 256, 256, 0, stream>>>(rowv, colv, deg, nrm, E);

    long long st = (long long)E * CH;
    scatter_kernel<<<(int)((st + 255) / 256), 256, 0, stream>>>(rowv, colv, nrm, mb,
                                                                agg, st);

    post_kernel<<<gemmGrid, 256, 0, stream>>>(agg, h1, og, Var, bar, Wgw, bgw,
                                              Wout, bout, out, nNodes, nTiles);
}